// EdgeGatedAttention_45913200394561
// MI455X (gfx1250) — compile-verified
//
#include <hip/hip_runtime.h>

// Edge-gated graph attention for MI455X (gfx1250, wave32).
// GEMM-shaped math except gate1@Wg2 (scalar fmaf) uses V_WMMA_F32_16X16X4_F32 (full fp32 precision).
// q/k/v (77MB) are L2-resident on MI455X (192MB L2) so edge gathers stay on-chip.

typedef __attribute__((ext_vector_type(2))) float v2f;
typedef __attribute__((ext_vector_type(8))) float v8f;

#define WAVES 8   // 8 waves * 32 lanes = 256 threads

// ---- order-preserving float<->uint mapping for atomic max on signed floats ----
__device__ __forceinline__ unsigned fkey(float f) {
  unsigned u = __float_as_uint(f);
  return (u & 0x80000000u) ? ~u : (u | 0x80000000u);
}
__device__ __forceinline__ float funkey(unsigned kk) {
  unsigned u = (kk & 0x80000000u) ? (kk & 0x7FFFFFFFu) : ~kk;
  return __uint_as_float(u);
}

// ============================================================================
// out[N x 128] = A[N x 128] @ W[128 x 128], fp32 WMMA 16x16x4.
// Block = 256 thr (8 waves); block handles 16 rows; wave w -> cols [16w,16w+16).
// ============================================================================
__global__ __launch_bounds__(256) void gemm128_kernel(
    const float* __restrict__ A, const float* __restrict__ W,
    float* __restrict__ out) {
  const int lane = threadIdx.x & 31;
  const int wv   = threadIdx.x >> 5;
  const int row0 = blockIdx.x << 4;
  const int n0   = wv << 4;
  const int m    = lane & 15;            // row (A-frag) / col (B,C frags)
  const int kk   = (lane >> 4) << 1;     // lanes 0-15: k+0,k+1 ; lanes 16-31: k+2,k+3
  const int mb   = (lane >> 4) << 3;     // C frag: rows 0-7 vs 8-15

  v8f c = {};
  const float* ar = A + (size_t)(row0 + m) * 128;
  const float* wp = W + n0 + m;
#pragma unroll 8
  for (int k0 = 0; k0 < 128; k0 += 4) {
    v2f a, b;
    a[0] = ar[k0 + kk];
    a[1] = ar[k0 + kk + 1];
    b[0] = wp[(size_t)(k0 + kk) * 128];
    b[1] = wp[(size_t)(k0 + kk + 1) * 128];
    c = __builtin_amdgcn_wmma_f32_16x16x4_f32(false, a, false, b, (short)0, c,
                                              false, false);
  }
  float* op = out + (size_t)(row0 + mb) * 128 + n0 + m;
#pragma unroll
  for (int r = 0; r < 8; ++r) op[(size_t)r * 128] = c[r];
}

// ============================================================================
// Pass 1 over edges: gate = silu(ea@Wg1)@Wg2 ; ek = ea@Wek ;
// scores[e][h] = dot(q[dst], k[src]+ek)/4 + gate[h] ; atomic segment-max.
// One wave per 16 edges. Head h == WMMA col-tile h (D_HEAD == 16).
// ============================================================================
__global__ __launch_bounds__(256) void edge_score_kernel(
    const float* __restrict__ ea, const int* __restrict__ src,
    const int* __restrict__ dst, const float* __restrict__ Wek,
    const float* __restrict__ Wg1, const float* __restrict__ Wg2,
    const float* __restrict__ q, const float* __restrict__ k,
    float* __restrict__ scores, unsigned* __restrict__ mmax) {
  __shared__ float g1s[WAVES][16][64];   // silu(ea@Wg1) per wave (32 KB)
  __shared__ float gts[WAVES][16][8];    // gate per wave

  const int lane = threadIdx.x & 31;
  const int wv   = threadIdx.x >> 5;
  const int e0   = (blockIdx.x * WAVES + wv) << 4;
  const int m    = lane & 15;
  const int kk   = (lane >> 4) << 1;
  const int mb   = (lane >> 4) << 3;

  // A fragments of the 16x32 edge_attr tile (reused for Wg1 and Wek GEMMs)
  v2f afr[8];
  const float* er = ea + (size_t)(e0 + m) * 32;
#pragma unroll
  for (int i = 0; i < 8; ++i) {
    afr[i][0] = er[4 * i + kk];
    afr[i][1] = er[4 * i + kk + 1];
  }

  // gate1 = silu(ea @ Wg1)  [16 x 64]
#pragma unroll
  for (int t = 0; t < 4; ++t) {
    v8f c = {};
    const float* wp = Wg1 + t * 16 + m;
#pragma unroll
    for (int i = 0; i < 8; ++i) {
      v2f b;
      b[0] = wp[(size_t)(4 * i + kk) * 64];
      b[1] = wp[(size_t)(4 * i + kk + 1) * 64];
      c = __builtin_amdgcn_wmma_f32_16x16x4_f32(false, afr[i], false, b,
                                                (short)0, c, false, false);
    }
#pragma unroll
    for (int r = 0; r < 8; ++r) {
      float xv = c[r];
      g1s[wv][mb + r][t * 16 + m] = xv / (1.0f + __expf(-xv));  // silu
    }
  }
  __syncthreads();

  // gate = gate1 @ Wg2  [16 x 8]; lane handles edge m, heads hb..hb+3
  {
    const int hb = (lane >> 4) << 2;
    float acc0 = 0.f, acc1 = 0.f, acc2 = 0.f, acc3 = 0.f;
#pragma unroll 4
    for (int j = 0; j < 64; ++j) {
      float g = g1s[wv][m][j];
      const float* w2 = Wg2 + j * 8 + hb;
      acc0 = fmaf(g, w2[0], acc0);
      acc1 = fmaf(g, w2[1], acc1);
      acc2 = fmaf(g, w2[2], acc2);
      acc3 = fmaf(g, w2[3], acc3);
    }
    gts[wv][m][hb + 0] = acc0;
    gts[wv][m][hb + 1] = acc1;
    gts[wv][m][hb + 2] = acc2;
    gts[wv][m][hb + 3] = acc3;
  }
  __syncthreads();

  int srcv[8], dstv[8];
#pragma unroll
  for (int r = 0; r < 8; ++r) {
    srcv[r] = src[e0 + mb + r];
    dstv[r] = dst[e0 + mb + r];
  }

#pragma unroll 1
  for (int h = 0; h < 8; ++h) {
    // ek tile for head h: ea @ Wek[:, 16h:16h+16]
    v8f c = {};
    const float* wp = Wek + h * 16 + m;
#pragma unroll
    for (int i = 0; i < 8; ++i) {
      v2f b;
      b[0] = wp[(size_t)(4 * i + kk) * 128];
      b[1] = wp[(size_t)(4 * i + kk + 1) * 128];
      c = __builtin_amdgcn_wmma_f32_16x16x4_f32(false, afr[i], false, b,
                                                (short)0, c, false, false);
    }
    const int col = h * 16 + m;
#pragma unroll
    for (int r = 0; r < 8; ++r) {
      float kv = k[(size_t)srcv[r] * 128 + col] + c[r];   // k_e (L2 gather)
      float qv = q[(size_t)dstv[r] * 128 + col];          // q_e (L2 gather)
      float p  = qv * kv;
      // reduce over the 16 lanes of each half-wave (head dim)
#pragma unroll
      for (int off = 8; off >= 1; off >>= 1) p += __shfl_xor(p, off, 32);
      if (m == 0) {
        float sc = p * 0.25f + gts[wv][mb + r][h];        // /sqrt(16) + gate
        scores[(size_t)(e0 + mb + r) * 8 + h] = sc;
        atomicMax(&mmax[(size_t)dstv[r] * 8 + h], fkey(sc));
      }
    }
  }
}

// ============================================================================
// Pass 2: exp_s = exp(score - m[dst]) stored in-place; denom += exp_s.
// ============================================================================
__global__ __launch_bounds__(256) void exp_denom_kernel(
    const int* __restrict__ dst, float* __restrict__ scores,
    const unsigned* __restrict__ mmax, float* __restrict__ denom, int total) {
  int idx = blockIdx.x * blockDim.x + threadIdx.x;
  if (idx >= total) return;
  int e = idx >> 3, h = idx & 7;
  int d = dst[e];
  float mv = funkey(mmax[(size_t)d * 8 + h]);
  float es = __expf(scores[idx] - mv);
  scores[idx] = es;
  atomicAdd(&denom[(size_t)d * 8 + h], es);
}

// ============================================================================
// Pass 3: v_e = v[src] + ea@Wev (recomputed, avoids 410MB spill);
// acc[dst] += (exp_s / (denom[dst]+1e-9)) * v_e  via fp32 L2 atomics.
// ============================================================================
__global__ __launch_bounds__(256) void edge_aggregate_kernel(
    const float* __restrict__ ea, const int* __restrict__ src,
    const int* __restrict__ dst, const float* __restrict__ Wev,
    const float* __restrict__ v, const float* __restrict__ scores,
    const float* __restrict__ denom, float* __restrict__ acc) {
  const int lane = threadIdx.x & 31;
  const int wv   = threadIdx.x >> 5;
  const int e0   = (blockIdx.x * WAVES + wv) << 4;
  const int m    = lane & 15;
  const int kk   = (lane >> 4) << 1;
  const int mb   = (lane >> 4) << 3;

  v2f afr[8];
  const float* er = ea + (size_t)(e0 + m) * 32;
#pragma unroll
  for (int i = 0; i < 8; ++i) {
    afr[i][0] = er[4 * i + kk];
    afr[i][1] = er[4 * i + kk + 1];
  }

  int srcv[8], dstv[8];
#pragma unroll
  for (int r = 0; r < 8; ++r) {
    srcv[r] = src[e0 + mb + r];
    dstv[r] = dst[e0 + mb + r];
  }

#pragma unroll 1
  for (int h = 0; h < 8; ++h) {
    v8f c = {};
    const float* wp = Wev + h * 16 + m;
#pragma unroll
    for (int i = 0; i < 8; ++i) {
      v2f b;
      b[0] = wp[(size_t)(4 * i + kk) * 128];
      b[1] = wp[(size_t)(4 * i + kk + 1) * 128];
      c = __builtin_amdgcn_wmma_f32_16x16x4_f32(false, afr[i], false, b,
                                                (short)0, c, false, false);
    }
    const int col = h * 16 + m;
#pragma unroll
    for (int r = 0; r < 8; ++r) {
      int e = e0 + mb + r;
      float att = scores[(size_t)e * 8 + h] /
                  (denom[(size_t)dstv[r] * 8 + h] + 1e-9f);
      float ve = v[(size_t)srcv[r] * 128 + col] + c[r];
      atomicAdd(&acc[(size_t)dstv[r] * 128 + col], att * ve);
    }
  }
}

// ============================================================================
extern "C" void kernel_launch(void* const* d_in, const int* in_sizes, int n_in,
                              void* d_out, int out_size, void* d_ws,
                              size_t ws_size, hipStream_t stream) {
  const float* x   = (const float*)d_in[0];
  const int*   ei  = (const int*)d_in[1];
  const float* ea  = (const float*)d_in[2];
  const float* Wq  = (const float*)d_in[3];
  const float* Wk  = (const float*)d_in[4];
  const float* Wv  = (const float*)d_in[5];
  const float* Wo  = (const float*)d_in[6];
  const float* Wek = (const float*)d_in[7];
  const float* Wev = (const float*)d_in[8];
  const float* Wg1 = (const float*)d_in[9];
  const float* Wg2 = (const float*)d_in[10];
  float* out = (float*)d_out;

  const int N = in_sizes[0] / 128;   // 50000
  const int E = in_sizes[2] / 32;    // 800000
  const int* src = ei;
  const int* dst = ei + E;

  // workspace layout (fp32): q | k | v | scores | acc | denom | mmax
  float* q      = (float*)d_ws;
  float* k      = q + (size_t)N * 128;
  float* v      = k + (size_t)N * 128;
  float* scores = v + (size_t)N * 128;
  float* acc    = scores + (size_t)E * 8;
  float* denom  = acc + (size_t)N * 128;
  unsigned* mmax = (unsigned*)(denom + (size_t)N * 8);

  // zero acc/denom/mmax (contiguous). fkey-space 0 == -inf sentinel.
  hipMemsetAsync(acc, 0, (size_t)(N * 128 + N * 8 + N * 8) * sizeof(float),
                 stream);

  dim3 blk(256);
  gemm128_kernel<<<dim3(N / 16), blk, 0, stream>>>(x, Wq, q);
  gemm128_kernel<<<dim3(N / 16), blk, 0, stream>>>(x, Wk, k);
  gemm128_kernel<<<dim3(N / 16), blk, 0, stream>>>(x, Wv, v);

  edge_score_kernel<<<dim3(E / (16 * WAVES)), blk, 0, stream>>>(
      ea, src, dst, Wek, Wg1, Wg2, q, k, scores, mmax);

  exp_denom_kernel<<<dim3((E * 8 + 255) / 256), blk, 0, stream>>>(
      dst, scores, mmax, denom, E * 8);

  edge_aggregate_kernel<<<dim3(E / (16 * WAVES)), blk, 0, stream>>>(
      ea, src, dst, Wev, v, scores, denom, acc);

  gemm128_kernel<<<dim3(N / 16), blk, 0, stream>>>(acc, Wo, out);
}